// Autoformer_34196529611188
// MI455X (gfx1250) — compile-verified
//
#include <hip/hip_runtime.h>
#include <cmath>
#include <cstddef>
#include <cstdint>

// ---------------------------------------------------------------------------
// Types for WMMA (CDNA5 / gfx1250, wave32)
// ---------------------------------------------------------------------------
typedef float v2f __attribute__((ext_vector_type(2)));
typedef float v8f __attribute__((ext_vector_type(8)));

#define N_HEADS 8
#define DM 512
#define DFF 2048
#define BSZ 32
#define LEN 512
#define LDEC 144
#define PRED 96
#define LABEL 48

// GEMM tile config: 256 threads = 8 waves; block tile 128x64, wave tile 32x32
// (2x2 grid of 16x16 WMMA accumulators per wave), K-block 32, double-buffered
// LDS filled via async global->LDS b128 copies (ASYNCcnt).
#define BM 128
#define BN 64
#define BK 32
#define APITCH 36   // 36 floats = 144B, multiple of 16B
#define BPITCH 68   // 68 floats = 272B, multiple of 16B

#define ASYNC_B128(ldsu32, gptr)                                              \
  asm volatile("global_load_async_to_lds_b128 %0, %1, off"                    \
               :: "v"(ldsu32), "v"(gptr) : "memory")
#define WAIT_ASYNC() asm volatile("s_wait_asynccnt 0x0" ::: "memory")

__device__ __forceinline__ unsigned lds_u32(const void* p) {
  // generic pointer to an LDS object: low 32 bits are the LDS byte offset
  return (unsigned)(uintptr_t)p;
}

// ---------------------------------------------------------------------------
// WMMA fp32 GEMM: C[M,N] = A[M,K] * W[K,N] (+bias) (opt ReLU)
// A row-major (lda=K), W row-major (ldw=N). M%128==0, N%64==0, K%32==0.
// ---------------------------------------------------------------------------
__global__ __launch_bounds__(256) void gemm_bias_kernel(
    const float* __restrict__ A, const float* __restrict__ W,
    const float* __restrict__ bias, float* __restrict__ C,
    int M, int N, int K, int relu)
{
  __shared__ float As[2][BM][APITCH];
  __shared__ float Bs[2][BK][BPITCH];

  const int tid  = threadIdx.x;
  const int wave = tid >> 5;
  const int lane = tid & 31;
  const int wm = (wave & 3) * 32;
  const int wn = (wave >> 2) * 32;
  const int half = lane >> 4;
  const int l16  = lane & 15;

  const size_t bm0 = (size_t)blockIdx.x * BM;
  const size_t bn0 = (size_t)blockIdx.y * BN;

  auto load_tiles = [&](int k0, int buf) {
    // A tile: 128x32 = 1024 b128 chunks, 4 per thread
#pragma unroll
    for (int u = 0; u < 4; ++u) {
      int id = u * 256 + tid;
      int r = id >> 3, c = (id & 7) * 4;
      unsigned l = lds_u32(&As[buf][r][c]);
      const float* g = A + (bm0 + r) * K + k0 + c;
      ASYNC_B128(l, g);
    }
    // B tile: 32x64 = 512 b128 chunks, 2 per thread
#pragma unroll
    for (int u = 0; u < 2; ++u) {
      int id = u * 256 + tid;
      int r = id >> 4, c = (id & 15) * 4;
      unsigned l = lds_u32(&Bs[buf][r][c]);
      const float* g = W + (size_t)(k0 + r) * N + bn0 + c;
      ASYNC_B128(l, g);
    }
  };

  v8f acc[2][2] = {};

  load_tiles(0, 0);
  WAIT_ASYNC();
  __syncthreads();

  int cur = 0;
  for (int k0 = 0; k0 < K; k0 += BK) {
    if (k0 + BK < K) load_tiles(k0 + BK, cur ^ 1);
#pragma unroll
    for (int kk = 0; kk < BK; kk += 4) {
      v2f a0, a1, b0, b1;
      a0.x = As[cur][wm + l16][kk + 2 * half + 0];
      a0.y = As[cur][wm + l16][kk + 2 * half + 1];
      a1.x = As[cur][wm + 16 + l16][kk + 2 * half + 0];
      a1.y = As[cur][wm + 16 + l16][kk + 2 * half + 1];
      b0.x = Bs[cur][kk + 2 * half + 0][wn + l16];
      b0.y = Bs[cur][kk + 2 * half + 1][wn + l16];
      b1.x = Bs[cur][kk + 2 * half + 0][wn + 16 + l16];
      b1.y = Bs[cur][kk + 2 * half + 1][wn + 16 + l16];
      acc[0][0] = __builtin_amdgcn_wmma_f32_16x16x4_f32(false, a0, false, b0, (short)0, acc[0][0], false, false);
      acc[0][1] = __builtin_amdgcn_wmma_f32_16x16x4_f32(false, a0, false, b1, (short)0, acc[0][1], false, false);
      acc[1][0] = __builtin_amdgcn_wmma_f32_16x16x4_f32(false, a1, false, b0, (short)0, acc[1][0], false, false);
      acc[1][1] = __builtin_amdgcn_wmma_f32_16x16x4_f32(false, a1, false, b1, (short)0, acc[1][1], false, false);
    }
    WAIT_ASYNC();
    __syncthreads();
    cur ^= 1;
  }

#pragma unroll
  for (int mi = 0; mi < 2; ++mi)
#pragma unroll
    for (int ni = 0; ni < 2; ++ni) {
      int col = (int)bn0 + wn + ni * 16 + l16;
      float bv = bias ? bias[col] : 0.f;
#pragma unroll
      for (int r = 0; r < 8; ++r) {
        int row = (int)bm0 + wm + mi * 16 + r + 8 * half;
        float v = acc[mi][ni][r] + bv;
        if (relu) v = fmaxf(v, 0.f);
        if (row < M && col < N) C[(size_t)row * N + col] = v;
      }
    }
}

// ---------------------------------------------------------------------------
// Batched S[b,i,j] = sum_d Q[b,i,d] * Km[b,j,d]  (B accessed transposed).
// B tile is staged in LDS as [n][k] so async b128 copies need no transpose.
// Guards handle L not multiple of tiles (L=144 decoder case).
// ---------------------------------------------------------------------------
__global__ __launch_bounds__(256) void gemm_qkt_kernel(
    const float* __restrict__ Q, const float* __restrict__ Km,
    float* __restrict__ S, int L, int D,
    long strideQ, long strideK)
{
  __shared__ float As[2][BM][APITCH];
  __shared__ float Bs[2][BN][APITCH];   // [n][k] layout, pitch 36

  const int b = blockIdx.z;
  const float* Ab = Q  + (size_t)b * strideQ;
  const float* Bb = Km + (size_t)b * strideK;
  float* Sb = S + (size_t)b * L * L;

  const int tid  = threadIdx.x;
  const int wave = tid >> 5;
  const int lane = tid & 31;
  const int wm = (wave & 3) * 32;
  const int wn = (wave >> 2) * 32;
  const int half = lane >> 4;
  const int l16  = lane & 15;

  const int bm0 = blockIdx.x * BM;
  const int bn0 = blockIdx.y * BN;

  auto load_tiles = [&](int k0, int buf) {
#pragma unroll
    for (int u = 0; u < 4; ++u) {            // A: 128x32
      int id = u * 256 + tid;
      int r = id >> 3, c = (id & 7) * 4;
      int gr = bm0 + r; if (gr >= L) gr = L - 1;
      unsigned l = lds_u32(&As[buf][r][c]);
      const float* g = Ab + (size_t)gr * D + k0 + c;
      ASYNC_B128(l, g);
    }
#pragma unroll
    for (int u = 0; u < 2; ++u) {            // B: 64 rows(n) x 32 cols(k)
      int id = u * 256 + tid;
      int r = id >> 3, c = (id & 7) * 4;
      int gn = bn0 + r; if (gn >= L) gn = L - 1;
      unsigned l = lds_u32(&Bs[buf][r][c]);
      const float* g = Bb + (size_t)gn * D + k0 + c;
      ASYNC_B128(l, g);
    }
  };

  v8f acc[2][2] = {};

  load_tiles(0, 0);
  WAIT_ASYNC();
  __syncthreads();

  int cur = 0;
  for (int k0 = 0; k0 < D; k0 += BK) {
    if (k0 + BK < D) load_tiles(k0 + BK, cur ^ 1);
#pragma unroll
    for (int kk = 0; kk < BK; kk += 4) {
      v2f a0, a1, b0, b1;
      a0.x = As[cur][wm + l16][kk + 2 * half + 0];
      a0.y = As[cur][wm + l16][kk + 2 * half + 1];
      a1.x = As[cur][wm + 16 + l16][kk + 2 * half + 0];
      a1.y = As[cur][wm + 16 + l16][kk + 2 * half + 1];
      b0.x = Bs[cur][wn + l16][kk + 2 * half + 0];
      b0.y = Bs[cur][wn + l16][kk + 2 * half + 1];
      b1.x = Bs[cur][wn + 16 + l16][kk + 2 * half + 0];
      b1.y = Bs[cur][wn + 16 + l16][kk + 2 * half + 1];
      acc[0][0] = __builtin_amdgcn_wmma_f32_16x16x4_f32(false, a0, false, b0, (short)0, acc[0][0], false, false);
      acc[0][1] = __builtin_amdgcn_wmma_f32_16x16x4_f32(false, a0, false, b1, (short)0, acc[0][1], false, false);
      acc[1][0] = __builtin_amdgcn_wmma_f32_16x16x4_f32(false, a1, false, b0, (short)0, acc[1][0], false, false);
      acc[1][1] = __builtin_amdgcn_wmma_f32_16x16x4_f32(false, a1, false, b1, (short)0, acc[1][1], false, false);
    }
    WAIT_ASYNC();
    __syncthreads();
    cur ^= 1;
  }

#pragma unroll
  for (int mi = 0; mi < 2; ++mi)
#pragma unroll
    for (int ni = 0; ni < 2; ++ni) {
      int col = bn0 + wn + ni * 16 + l16;
#pragma unroll
      for (int r = 0; r < 8; ++r) {
        int row = bm0 + wm + mi * 16 + r + 8 * half;
        if (row < L && col < L) Sb[(size_t)row * L + col] = acc[mi][ni][r];
      }
    }
}

// ---------------------------------------------------------------------------
// mean_corr[b,tau] = (1/(H*E)) * sum_t S[b, (t+tau)%L, t]
// ---------------------------------------------------------------------------
__global__ void diag_corr_kernel(const float* __restrict__ S,
                                 float* __restrict__ mc, int Bn, int L, float inv)
{
  int idx = blockIdx.x * blockDim.x + threadIdx.x;
  if (idx >= Bn * L) return;
  int b = idx / L, tau = idx % L;
  const float* Sb = S + (size_t)b * L * L;
  float s = 0.f;
  for (int t = 0; t < L; ++t) {
    int r = t + tau; if (r >= L) r -= L;
    s += Sb[(size_t)r * L + t];
  }
  mc[idx] = s * inv;
}

// ---------------------------------------------------------------------------
// Per-batch top-k (k<=6) + softmax weights.
// ---------------------------------------------------------------------------
__global__ __launch_bounds__(256) void topk_softmax_kernel(
    const float* __restrict__ mc, float* __restrict__ w,
    int* __restrict__ delay, int L, int topk)
{
  __shared__ float sv[256];
  __shared__ int   si[256];
  __shared__ int   chosen[8];
  __shared__ float cw[8];
  const int b = blockIdx.x;
  const float* m = mc + (size_t)b * L;
  const int tid = threadIdx.x;

  for (int it = 0; it < topk; ++it) {
    float best = -3.4e38f; int bi = -1;
    for (int i = tid; i < L; i += 256) {
      bool skip = false;
      for (int j = 0; j < it; ++j) if (chosen[j] == i) skip = true;
      float v = m[i];
      if (!skip && v > best) { best = v; bi = i; }
    }
    sv[tid] = best; si[tid] = bi;
    __syncthreads();
    for (int s = 128; s > 0; s >>= 1) {
      if (tid < s && sv[tid + s] > sv[tid]) { sv[tid] = sv[tid + s]; si[tid] = si[tid + s]; }
      __syncthreads();
    }
    if (tid == 0) { chosen[it] = si[0]; cw[it] = sv[0]; }
    __syncthreads();
  }
  if (tid == 0) {
    float mx = cw[0];
    for (int j = 1; j < topk; ++j) mx = fmaxf(mx, cw[j]);
    float s = 0.f;
    for (int j = 0; j < topk; ++j) { float e = __expf(cw[j] - mx); cw[j] = e; s += e; }
    for (int j = 0; j < topk; ++j) {
      w[b * 8 + j] = cw[j] / s;
      delay[b * 8 + j] = chosen[j];
    }
  }
}

// ---------------------------------------------------------------------------
// out[b,t,d] = sum_i w[b,i] * V[b, (t+delay[b,i])%L, d]
// ---------------------------------------------------------------------------
__global__ void agg_kernel(const float* __restrict__ V, const float* __restrict__ w,
                           const int* __restrict__ delay, float* __restrict__ out,
                           int Bn, int L, int D, long strideV, int topk)
{
  size_t idx = (size_t)blockIdx.x * blockDim.x + threadIdx.x;
  size_t n = (size_t)Bn * L * D;
  if (idx >= n) return;
  int d = (int)(idx % D);
  size_t row = idx / D;
  int t = (int)(row % L);
  int b = (int)(row / L);
  const float* Vb = V + (size_t)b * strideV;
  float s = 0.f;
  for (int i = 0; i < topk; ++i) {
    int tt = t + delay[b * 8 + i]; if (tt >= L) tt -= L;
    s += w[b * 8 + i] * Vb[(size_t)tt * D + d];
  }
  out[idx] = s;
}

// ---------------------------------------------------------------------------
// series_decomp: moving mean (k=25, edge replicate). seas = x - m, trend (+)= m
// ---------------------------------------------------------------------------
__global__ void decomp_kernel(const float* __restrict__ x, float* __restrict__ seas,
                              float* __restrict__ trend, int rows, int L, int D, int acc)
{
  size_t idx = (size_t)blockIdx.x * blockDim.x + threadIdx.x;
  size_t n = (size_t)rows * D;
  if (idx >= n) return;
  int d = (int)(idx % D);
  size_t row = idx / D;
  int l = (int)(row % L);
  size_t b = row / L;
  const float* base = x + (b * L) * (size_t)D + d;
  float s = 0.f;
#pragma unroll
  for (int j = -12; j <= 12; ++j) {
    int ll = l + j;
    ll = ll < 0 ? 0 : (ll >= L ? L - 1 : ll);
    s += base[(size_t)ll * D];
  }
  float m = s * (1.0f / 25.0f);
  if (seas)  seas[idx] = x[idx] - m;
  if (trend) { if (acc) trend[idx] += m; else trend[idx] = m; }
}

// ---------------------------------------------------------------------------
// embed: circular conv3 over channels + time-mark linear
// ---------------------------------------------------------------------------
__global__ void embed_kernel(const float* __restrict__ x, const float* __restrict__ xmark,
                             const float* __restrict__ tokW, const float* __restrict__ timeW,
                             float* __restrict__ out, int Bn, int L, int C, int Mk, int D)
{
  size_t idx = (size_t)blockIdx.x * blockDim.x + threadIdx.x;
  size_t n = (size_t)Bn * L * D;
  if (idx >= n) return;
  int o = (int)(idx % D);
  size_t row = idx / D;
  int l = (int)(row % L);
  int b = (int)(row / L);
  float s = 0.f;
  for (int j = 0; j < 3; ++j) {
    int ll = l + j - 1; if (ll < 0) ll += L; if (ll >= L) ll -= L;
    const float* xr = x + ((size_t)b * L + ll) * C;
    for (int c = 0; c < C; ++c) s += xr[c] * tokW[((size_t)o * C + c) * 3 + j];
  }
  const float* mr = xmark + ((size_t)b * L + l) * Mk;
  for (int m = 0; m < Mk; ++m) s += mr[m] * timeW[(size_t)m * D + o];
  out[idx] = s;
}

// ---------------------------------------------------------------------------
// layernorm over D (one row per block), then separate column de-mean.
// ---------------------------------------------------------------------------
__global__ __launch_bounds__(256) void layernorm_kernel(
    const float* __restrict__ x, const float* __restrict__ g,
    const float* __restrict__ bb, float* __restrict__ y, int D)
{
  __shared__ float red[256];
  const int row = blockIdx.x;
  const int tid = threadIdx.x;
  const float* xr = x + (size_t)row * D;
  float s = 0.f;
  for (int d = tid; d < D; d += 256) s += xr[d];
  red[tid] = s; __syncthreads();
  for (int st = 128; st > 0; st >>= 1) { if (tid < st) red[tid] += red[tid + st]; __syncthreads(); }
  float mu = red[0] / D; __syncthreads();
  float v = 0.f;
  for (int d = tid; d < D; d += 256) { float t = xr[d] - mu; v += t * t; }
  red[tid] = v; __syncthreads();
  for (int st = 128; st > 0; st >>= 1) { if (tid < st) red[tid] += red[tid + st]; __syncthreads(); }
  float inv = rsqrtf(red[0] / D + 1e-5f);
  for (int d = tid; d < D; d += 256)
    y[(size_t)row * D + d] = (xr[d] - mu) * inv * g[d] + bb[d];
}

__global__ void colmean_sub_kernel(float* __restrict__ y, int Bn, int L, int D)
{
  int idx = blockIdx.x * blockDim.x + threadIdx.x;
  if (idx >= Bn * D) return;
  int b = idx / D, d = idx % D;
  float* base = y + ((size_t)b * L) * D + d;
  float s = 0.f;
  for (int l = 0; l < L; ++l) s += base[(size_t)l * D];
  s /= L;
  for (int l = 0; l < L; ++l) base[(size_t)l * D] -= s;
}

// ---------------------------------------------------------------------------
// misc elementwise
// ---------------------------------------------------------------------------
__global__ void add_kernel(const float* __restrict__ a, const float* __restrict__ b,
                           float* __restrict__ c, size_t n)
{
  size_t idx = (size_t)blockIdx.x * blockDim.x + threadIdx.x;
  if (idx < n) c[idx] = a[idx] + b[idx];
}

__global__ void enc_mean_kernel(const float* __restrict__ x, float* __restrict__ mean, int Bn)
{
  int idx = blockIdx.x * blockDim.x + threadIdx.x;
  if (idx >= Bn * 8) return;
  int b = idx / 8, c = idx % 8;
  float s = 0.f;
  for (int l = 0; l < LEN; ++l) s += x[((size_t)b * LEN + l) * 8 + c];
  mean[idx] = s / (float)LEN;
}

__global__ void build_init_kernel(const float* __restrict__ seasS, const float* __restrict__ trendS,
                                  const float* __restrict__ encmean,
                                  const float* __restrict__ xmark_enc, const float* __restrict__ xmark_dec,
                                  float* __restrict__ seas_init, float* __restrict__ trend_init,
                                  float* __restrict__ markdec, int Bn)
{
  int idx = blockIdx.x * blockDim.x + threadIdx.x;
  if (idx >= Bn * LDEC) return;
  int b = idx / LDEC, l = idx % LDEC;
  for (int c = 0; c < 8; ++c) {
    float tv, sv;
    if (l < LABEL) {
      size_t src = ((size_t)b * LEN + (LEN - LABEL) + l) * 8 + c;
      tv = trendS[src]; sv = seasS[src];
    } else { tv = encmean[b * 8 + c]; sv = 0.f; }
    trend_init[((size_t)b * LDEC + l) * 8 + c] = tv;
    seas_init [((size_t)b * LDEC + l) * 8 + c] = sv;
  }
  for (int m = 0; m < 4; ++m) {
    float mv = (l < LABEL)
      ? xmark_enc[((size_t)b * LEN + (LEN - LABEL) + l) * 4 + m]
      : xmark_dec[((size_t)b * PRED + (l - LABEL)) * 4 + m];
    markdec[((size_t)b * LDEC + l) * 4 + m] = mv;
  }
}

// rt[b,l,c] = circular conv3 of trend-sum (B,LDEC,DM) with Wtrend (8,DM,3)
__global__ void trend_conv_kernel(const float* __restrict__ ts, const float* __restrict__ Wt,
                                  float* __restrict__ rt, int Bn)
{
  int idx = blockIdx.x * blockDim.x + threadIdx.x;
  if (idx >= Bn * LDEC * 8) return;
  int c = idx % 8;
  int row = idx / 8;
  int l = row % LDEC;
  int b = row / LDEC;
  float s = 0.f;
  for (int j = 0; j < 3; ++j) {
    int ll = l + j - 1; if (ll < 0) ll += LDEC; if (ll >= LDEC) ll -= LDEC;
    const float* tr = ts + ((size_t)b * LDEC + ll) * DM;
    const float* wc = Wt + (size_t)c * DM * 3 + j;
    for (int d = 0; d < DM; ++d) s += tr[d] * wc[(size_t)d * 3];
  }
  rt[idx] = s;
}

// seasonal projection: N=8 (too thin for WMMA tiles)
__global__ void proj_kernel(const float* __restrict__ x, const float* __restrict__ W,
                            const float* __restrict__ bias, float* __restrict__ out,
                            int rows, int K, int N)
{
  int idx = blockIdx.x * blockDim.x + threadIdx.x;
  if (idx >= rows * N) return;
  int r = idx / N, nn = idx % N;
  const float* xr = x + (size_t)r * K;
  float s = bias[nn];
  for (int k = 0; k < K; ++k) s += xr[k] * W[(size_t)k * N + nn];
  out[idx] = s;
}

__global__ void final_kernel(const float* __restrict__ trend, const float* __restrict__ seas,
                             float* __restrict__ out, int Bn)
{
  int idx = blockIdx.x * blockDim.x + threadIdx.x;
  if (idx >= Bn * PRED * 8) return;
  int c = idx % 8;
  int r = idx / 8;
  int l = r % PRED;
  int b = r / PRED;
  size_t src = ((size_t)b * LDEC + LABEL + l) * 8 + c;
  out[idx] = trend[src] + seas[src];
}

// ---------------------------------------------------------------------------
// Host orchestration
// ---------------------------------------------------------------------------
struct AttnP { const float *Wq,*bq,*Wk,*bk,*Wv,*bv,*Wo,*bo; };

static inline int ceil_div(int a, int b) { return (a + b - 1) / b; }

extern "C" void kernel_launch(void* const* d_in, const int* in_sizes, int n_in,
                              void* d_out, int out_size, void* d_ws, size_t ws_size,
                              hipStream_t stream)
{
  (void)in_sizes; (void)n_in; (void)out_size; (void)ws_size;
  const float* x_enc      = (const float*)d_in[0];
  const float* x_mark_enc = (const float*)d_in[1];
  // d_in[2] = x_dec (unused by reference math beyond shape)
  const float* x_mark_dec = (const float*)d_in[3];

  int pi = 4;
  auto nxt = [&]() { return (const float*)d_in[pi++]; };
  auto nxtAttn = [&]() {
    AttnP a;
    a.Wq = nxt(); a.bq = nxt(); a.Wk = nxt(); a.bk = nxt();
    a.Wv = nxt(); a.bv = nxt(); a.Wo = nxt(); a.bo = nxt();
    return a;
  };

  const float* enc_tok_W  = nxt();
  const float* enc_time_W = nxt();
  const float* dec_tok_W  = nxt();
  const float* dec_time_W = nxt();
  AttnP encA[2]; const float *encWc1[2], *encWc2[2];
  for (int i = 0; i < 2; ++i) { encA[i] = nxtAttn(); encWc1[i] = nxt(); encWc2[i] = nxt(); }
  const float* enc_g = nxt();
  const float* enc_b = nxt();
  AttnP selfA = nxtAttn();
  AttnP crossA = nxtAttn();
  const float* dWc1   = nxt();
  const float* dWc2   = nxt();
  const float* Wtrend = nxt();
  const float* dec_g  = nxt();
  const float* dec_b  = nxt();
  const float* proj_W = nxt();
  const float* proj_b = nxt();

  // ---- workspace arena (floats) ----
  const size_t SZ_ENC = (size_t)BSZ * LEN * DM;   // 8.39M
  const size_t SZ_DEC = (size_t)BSZ * LDEC * DM;  // 2.36M
  float* ws = (float*)d_ws;
  size_t off = 0;
  auto alloc = [&](size_t n) { float* p = ws + off; off += n; return p; };

  float* X      = alloc(SZ_ENC);
  float* Q      = alloc(SZ_ENC);   // H (FFN hidden, 4*SZ_ENC) aliases Q..Sc
  float* Kb     = alloc(SZ_ENC);
  float* Vb     = alloc(SZ_ENC);
  float* Sc     = alloc(SZ_ENC);
  float* T0     = alloc(SZ_ENC);
  float* T1     = alloc(SZ_ENC);
  float* ENCOUT = alloc(SZ_ENC);
  float* H      = Q; // alias: FFN hidden needs 4*SZ_ENC == Q+Kb+Vb+Sc span
  float* XD     = alloc(SZ_DEC);
  float* TSUM   = alloc(SZ_DEC);
  float* MC     = alloc((size_t)BSZ * LEN);
  float* WTOP   = alloc(BSZ * 8);
  int*   DTOP   = (int*)alloc(BSZ * 8);
  float* ENCM   = alloc(BSZ * 8);
  float* TREND  = alloc((size_t)BSZ * LDEC * 8);
  float* SEASI  = alloc((size_t)BSZ * LDEC * 8);
  float* MARKD  = alloc((size_t)BSZ * LDEC * 4);
  float* SEASS  = alloc((size_t)BSZ * LEN * 8);
  float* TRENDS = alloc((size_t)BSZ * LEN * 8);
  float* RT     = alloc((size_t)BSZ * LDEC * 8);
  float* PROJ   = alloc((size_t)BSZ * LDEC * 8);

  auto gemm = [&](const float* A, const float* W, const float* bias, float* C,
                  int M, int N, int K, int relu) {
    dim3 g(M / BM, N / BN);
    gemm_bias_kernel<<<g, 256, 0, stream>>>(A, W, bias, C, M, N, K, relu);
  };
  auto ew = [&](size_t n) { return (unsigned)((n + 255) / 256); };

  // attention block: out = AutoCorr(x_q W_q, x_kv W_k, x_kv W_v) W_o
  auto attn = [&](const float* xq, int Lq, const float* xkv, int Skv,
                  const AttnP& p, float* outb) {
    const int Mq = BSZ * Lq, Mkv = BSZ * Skv;
    gemm(xq,  p.Wq, p.bq, Q,  Mq,  DM, DM, 0);
    gemm(xkv, p.Wk, p.bk, Kb, Mkv, DM, DM, 0);
    gemm(xkv, p.Wv, p.bv, Vb, Mkv, DM, DM, 0);
    dim3 gq(ceil_div(Lq, BM), ceil_div(Lq, BN), BSZ);
    gemm_qkt_kernel<<<gq, 256, 0, stream>>>(Q, Kb, Sc, Lq, DM,
                                            (long)Lq * DM, (long)Skv * DM);
    diag_corr_kernel<<<ew((size_t)BSZ * Lq), 256, 0, stream>>>(
        Sc, MC, BSZ, Lq, 1.0f / (float)DM);
    int topk = (int)floor(log((double)Lq));   // 6 for L=512, 4 for L=144
    topk_softmax_kernel<<<BSZ, 256, 0, stream>>>(MC, WTOP, DTOP, Lq, topk);
    agg_kernel<<<ew((size_t)Mq * DM), 256, 0, stream>>>(
        Vb, WTOP, DTOP, T0, BSZ, Lq, DM, (long)Skv * DM, topk);
    gemm(T0, p.Wo, p.bo, outb, Mq, DM, DM, 0);
  };

  // ======================= encoder =======================
  embed_kernel<<<ew(SZ_ENC), 256, 0, stream>>>(
      x_enc, x_mark_enc, enc_tok_W, enc_time_W, X, BSZ, LEN, 8, 4, DM);

  for (int layer = 0; layer < 2; ++layer) {
    attn(X, LEN, X, LEN, encA[layer], T1);
    add_kernel<<<ew(SZ_ENC), 256, 0, stream>>>(X, T1, T0, SZ_ENC);
    decomp_kernel<<<ew(SZ_ENC), 256, 0, stream>>>(T0, X, nullptr, BSZ * LEN, LEN, DM, 0);
    gemm(X, encWc1[layer], nullptr, H, BSZ * LEN, DFF, DM, 1);
    gemm(H, encWc2[layer], nullptr, T1, BSZ * LEN, DM, DFF, 0);
    add_kernel<<<ew(SZ_ENC), 256, 0, stream>>>(X, T1, T0, SZ_ENC);
    decomp_kernel<<<ew(SZ_ENC), 256, 0, stream>>>(T0, X, nullptr, BSZ * LEN, LEN, DM, 0);
  }
  layernorm_kernel<<<BSZ * LEN, 256, 0, stream>>>(X, enc_g, enc_b, ENCOUT, DM);
  colmean_sub_kernel<<<ew(BSZ * DM), 256, 0, stream>>>(ENCOUT, BSZ, LEN, DM);

  // ======================= decoder init =======================
  decomp_kernel<<<ew((size_t)BSZ * LEN * 8), 256, 0, stream>>>(
      x_enc, SEASS, TRENDS, BSZ * LEN, LEN, 8, 0);
  enc_mean_kernel<<<ew(BSZ * 8), 256, 0, stream>>>(x_enc, ENCM, BSZ);
  build_init_kernel<<<ew(BSZ * LDEC), 256, 0, stream>>>(
      SEASS, TRENDS, ENCM, x_mark_enc, x_mark_dec, SEASI, TREND, MARKD, BSZ);
  embed_kernel<<<ew(SZ_DEC), 256, 0, stream>>>(
      SEASI, MARKD, dec_tok_W, dec_time_W, XD, BSZ, LDEC, 8, 4, DM);

  // ======================= decoder layer (DL=1) =======================
  attn(XD, LDEC, XD, LDEC, selfA, T1);
  add_kernel<<<ew(SZ_DEC), 256, 0, stream>>>(XD, T1, T0, SZ_DEC);
  decomp_kernel<<<ew(SZ_DEC), 256, 0, stream>>>(T0, XD, TSUM, BSZ * LDEC, LDEC, DM, 0);

  attn(XD, LDEC, ENCOUT, LEN, crossA, T1);
  add_kernel<<<ew(SZ_DEC), 256, 0, stream>>>(XD, T1, T0, SZ_DEC);
  decomp_kernel<<<ew(SZ_DEC), 256, 0, stream>>>(T0, XD, TSUM, BSZ * LDEC, LDEC, DM, 1);

  gemm(XD, dWc1, nullptr, H, BSZ * LDEC, DFF, DM, 1);
  gemm(H, dWc2, nullptr, T1, BSZ * LDEC, DM, DFF, 0);
  add_kernel<<<ew(SZ_DEC), 256, 0, stream>>>(XD, T1, T0, SZ_DEC);
  decomp_kernel<<<ew(SZ_DEC), 256, 0, stream>>>(T0, XD, TSUM, BSZ * LDEC, LDEC, DM, 1);

  trend_conv_kernel<<<ew((size_t)BSZ * LDEC * 8), 256, 0, stream>>>(TSUM, Wtrend, RT, BSZ);
  add_kernel<<<ew((size_t)BSZ * LDEC * 8), 256, 0, stream>>>(
      TREND, RT, TREND, (size_t)BSZ * LDEC * 8);

  layernorm_kernel<<<BSZ * LDEC, 256, 0, stream>>>(XD, dec_g, dec_b, T1, DM);
  colmean_sub_kernel<<<ew(BSZ * DM), 256, 0, stream>>>(T1, BSZ, LDEC, DM);
  proj_kernel<<<ew((size_t)BSZ * LDEC * 8), 256, 0, stream>>>(
      T1, proj_W, proj_b, PROJ, BSZ * LDEC, DM, 8);

  final_kernel<<<ew((size_t)BSZ * PRED * 8), 256, 0, stream>>>(
      TREND, PROJ, (float*)d_out, BSZ);
}